// NMRRenderer_31456340476207
// MI455X (gfx1250) — compile-verified
//
#include <hip/hip_runtime.h>
#include <math.h>

#define IMG   256
#define PIX   (IMG*IMG)
#define NEARP 0.1f
#define FARP  100.0f
#define INVFAR 0.01f      // 1/FAR: score of "no hit"
#define NFACE 13776
#define NB    2
#define NVERT 6890
#define REC   16          // floats per face record (64 bytes)
#define SC    256         // faces per LDS superchunk (multiple of 16)
#define NSUP  ((NFACE + SC - 1)/SC)

typedef float v2f __attribute__((ext_vector_type(2)));
typedef float v8f __attribute__((ext_vector_type(8)));
typedef int   v4i __attribute__((vector_size(16)));   // matches builtin param type

// ---- CDNA5 async global->LDS path (guarded; falls back to sync loads) ----
#ifndef USE_ASYNC_LDS
#if defined(__has_builtin)
#if __has_builtin(__builtin_amdgcn_global_load_async_to_lds_b128)
#define USE_ASYNC_LDS 1
#endif
#endif
#endif
#ifndef USE_ASYNC_LDS
#define USE_ASYNC_LDS 0
#endif

#ifndef HAVE_WAIT_ASYNC
#if defined(__has_builtin)
#if __has_builtin(__builtin_amdgcn_s_wait_asynccnt)
#define HAVE_WAIT_ASYNC 1
#endif
#endif
#endif
#ifndef HAVE_WAIT_ASYNC
#define HAVE_WAIT_ASYNC 0
#endif

__device__ __forceinline__ void wait_async_all() {
#if USE_ASYNC_LDS
#if HAVE_WAIT_ASYNC
  __builtin_amdgcn_s_wait_asynccnt(0);
#else
  asm volatile("s_wait_asynccnt 0" ::: "memory");
#endif
#endif
}

__device__ __forceinline__ void stage_chunk(const float* __restrict__ gsrc,
                                            float* ldst, int n4, int tid) {
#if USE_ASYNC_LDS
  typedef __attribute__((address_space(1))) v4i* gptr_t;   // global int4*
  typedef __attribute__((address_space(3))) v4i* lptr_t;   // LDS int4*
  for (int i = tid; i < n4; i += 256) {
    gptr_t g = (gptr_t)(unsigned long long)(const void*)(gsrc + i*4);
    lptr_t l = (lptr_t)(unsigned)(unsigned long long)(const void*)(ldst + i*4);
    __builtin_amdgcn_global_load_async_to_lds_b128(g, l, 0, 0);
  }
#else
  const float4* __restrict__ g4 = (const float4*)gsrc;
  float4* l4 = (float4*)ldst;
  for (int i = tid; i < n4; i += 256) l4[i] = g4[i];
#endif
}

// ---- Pass 1: project vertices, build per-face affine edge-function records ----
// record: [a0,b0,c0,0][a1,b1,c1,0][az,bz,cz,0][z0s,z1s,z2s,0]
__global__ __launch_bounds__(256) void nmr_setup(
    const float* __restrict__ verts, const float* __restrict__ cam_ts,
    const int*   __restrict__ faces, const float* __restrict__ camK,
    const float* __restrict__ camR,  float* __restrict__ coef)
{
  int idx = blockIdx.x*blockDim.x + threadIdx.x;
  if (idx >= NB*NFACE) return;
  int b = idx / NFACE, f = idx % NFACE;
  const float* R = camR + b*9;
  const float* K = camK + b*9;
  const float* t = cam_ts + b*3;
  float X[3], Y[3], Z[3];
#pragma unroll
  for (int c = 0; c < 3; ++c) {
    int vi = faces[f*3 + c];
    const float* v = verts + ((size_t)b*NVERT + vi)*3;
    float vx = R[0]*v[0] + R[1]*v[1] + R[2]*v[2] + t[0];
    float vy = R[3]*v[0] + R[4]*v[1] + R[5]*v[2] + t[1];
    float vz = R[6]*v[0] + R[7]*v[1] + R[8]*v[2] + t[2];
    float x  = vx / (vz + 1e-9f);
    float y  = vy / (vz + 1e-9f);
    float p0 = K[0]*x + K[1]*y + K[2];
    float p1 = K[3]*x + K[4]*y + K[5];
    X[c] = 2.0f*(p0 - IMG*0.5f)/(float)IMG;
    Y[c] = 2.0f*((float)IMG - p1 - IMG*0.5f)/(float)IMG;
    Z[c] = vz;
  }
  float x0=X[0],x1=X[1],x2=X[2], y0=Y[0],y1=Y[1],y2=Y[2], z0=Z[0],z1=Z[1],z2=Z[2];
  float den  = (y1-y2)*(x0-x2) + (x2-x1)*(y0-y2);
  bool  good = fabsf(den) > 1e-10f;
  float inv  = 1.0f / (good ? den : 1.0f);
  bool  zok  = (z0 > NEARP) && (z1 > NEARP) && (z2 > NEARP);
  float z0s = z0 > NEARP ? z0 : 1.0f;
  float z1s = z1 > NEARP ? z1 : 1.0f;
  float z2s = z2 > NEARP ? z2 : 1.0f;
  float a0 = (y1-y2)*inv, b0 = (x2-x1)*inv;
  float c0 = -(a0*x2 + b0*y2);
  float a1 = (y2-y0)*inv, b1 = (x0-x2)*inv;
  float c1 = -(a1*x2 + b1*y2);
  float r0 = 1.0f/z0s, r1 = 1.0f/z1s, r2 = 1.0f/z2s;
  float az = a0*(r0-r2) + a1*(r1-r2);
  float bz = b0*(r0-r2) + b1*(r1-r2);
  float cz = c0*(r0-r2) + c1*(r1-r2) + r2;
  if (!(good && zok)) {  // force "outside" so the face can never win
    a0 = 0.f; b0 = 0.f; c0 = -1.0f;
    a1 = 0.f; b1 = 0.f; c1 = 0.f;
    az = 0.f; bz = 0.f; cz = 0.f;
  }
  float4* o = (float4*)(coef + (size_t)idx*REC);
  o[0] = make_float4(a0, b0, c0, 0.f);
  o[1] = make_float4(a1, b1, c1, 0.f);
  o[2] = make_float4(az, bz, cz, 0.f);
  o[3] = make_float4(z0s, z1s, z2s, 0.f);
}

// ---- Pass 2: WMMA rasterize + shade. One wave32 owns 16 pixels. ----
__global__ __launch_bounds__(256) void nmr_raster(
    const float* __restrict__ coef, const float* __restrict__ tex,
    const float* __restrict__ parts3d, int* __restrict__ out)
{
  __shared__ float sh[2][SC*REC];   // 32 KB double buffer
  const int tid  = threadIdx.x;
  const int lane = tid & 31, wave = tid >> 5;
  const int n    = lane & 15, half = lane >> 4;
  const int b    = blockIdx.y;
  const int p    = blockIdx.x*128 + wave*16 + n;
  const float px = (2.0f*(float)(p & (IMG-1)) + 1.0f)/(float)IMG - 1.0f;
  const float py = (2.0f*(float)(p >> 8)      + 1.0f)/(float)IMG - 1.0f;

  // B matrix (4xK x 16 pixels): lanes 0-15 supply K=0,1 = (px,py); lanes 16-31 K=2,3 = (1,0)
  v2f bmat;
  bmat.x = half ? 1.0f : px;
  bmat.y = half ? 0.0f : py;
  const v8f zero = {0.f,0.f,0.f,0.f,0.f,0.f,0.f,0.f};

  // Track score = clamped 1/depth (monotonic: argmin depth == argmax score).
  // Strict > update keeps the earliest face at the max, reproducing the
  // reference's chunked argmin-first + strict-< zbuffer scan order.
  float bs = INVFAR;
  int   bm = -1;
  const float* cb = coef + (size_t)b*NFACE*REC;

  stage_chunk(cb, sh[0], min(SC, NFACE)*4, tid);
  wait_async_all();
  __syncthreads();

  for (int s = 0; s < NSUP; ++s) {
    if (s + 1 < NSUP) {
      int nfn = min(SC, NFACE - (s+1)*SC);
      stage_chunk(cb + (size_t)(s+1)*SC*REC, sh[(s+1)&1], nfn*4, tid);
    }
    const float* buf = sh[s & 1];
    const int nf = min(SC, NFACE - s*SC);     // always a multiple of 16
    const int nc = nf >> 4;
    const int fb = s*SC + half*8;
    for (int c = 0; c < nc; ++c) {
      const float* rec = buf + (c*16 + n)*REC;
      // A operand: lanes 0-15 carry [a,b] (K=0,1); lanes 16-31 carry [c,0] (K=2,3)
      v2f A0 = *(const v2f*)(rec + 0 + half*2);
      v2f A1 = *(const v2f*)(rec + 4 + half*2);
      v2f AZ = *(const v2f*)(rec + 8 + half*2);
      v8f d0 = __builtin_amdgcn_wmma_f32_16x16x4_f32(false, A0, false, bmat, (short)0, zero, false, false);
      v8f d1 = __builtin_amdgcn_wmma_f32_16x16x4_f32(false, A1, false, bmat, (short)0, zero, false, false);
      v8f dz = __builtin_amdgcn_wmma_f32_16x16x4_f32(false, AZ, false, bmat, (short)0, zero, false, false);
      const int mb = fb + c*16;   // D rows: lanes<16 -> M=0..7, lanes>=16 -> M=8..15
#pragma unroll
      for (int r = 0; r < 8; ++r) {
        float w0 = d0[r], w1 = d1[r];
        float w2 = 1.0f - w0 - w1;
        bool inside = (w0 >= 0.0f) && (w1 >= 0.0f) && (w2 >= 0.0f);
        float sc = inside ? fmaxf(dz[r], 1e-6f) : INVFAR;  // branchless, no division
        bool take = sc > bs;
        bs = take ? sc : bs;
        bm = take ? (mb + r) : bm;
      }
    }
    wait_async_all();   // next chunk landed in LDS
    __syncthreads();    // everyone done reading buf before it is restaged
  }

  // merge the two M-halves per pixel; (score desc, face index asc) lex order
  float os = __shfl_xor(bs, 16, 32);
  int   om = __shfl_xor(bm, 16, 32);
  if (os > bs || (os == bs && om < bm)) { bs = os; bm = om; }

  if (half == 0) {
    int part = 0;
    if (bm >= 0) {
      float bd = 1.0f / bs;            // exact division once: reference's stored depth
      const float* rec = cb + (size_t)bm*REC;
      float w0 = rec[0]*px + rec[1]*py + rec[2];
      float w1 = rec[4]*px + rec[5]*py + rec[6];
      float w2 = 1.0f - w0 - w1;
      float c0 = fminf(fmaxf(w0*bd/rec[12], 0.f), 1.f);
      float c1 = fminf(fmaxf(w1*bd/rec[13], 0.f), 1.f);
      float c2 = fminf(fmaxf(w2*bd/rec[14], 0.f), 1.f);
      const float* tb = tex + ((size_t)b*NFACE + bm)*24;
      float rr = 0.f, gg = 0.f, bb = 0.f;
#pragma unroll
      for (int i = 0; i < 2; ++i)
#pragma unroll
        for (int j = 0; j < 2; ++j)
#pragma unroll
          for (int k = 0; k < 2; ++k) {
            float wc = (i ? c0 : 1.f-c0)*(j ? c1 : 1.f-c1)*(k ? c2 : 1.f-c2);
            const float* tp = tb + ((i*2+j)*2+k)*3;
            rr += wc*tp[0]; gg += wc*tp[1]; bb += wc*tp[2];
          }
      int p0 = (int)fminf(fmaxf(floorf(100.f*rr), 0.f), 99.f);
      int p1 = (int)fminf(fmaxf(floorf(100.f*gg), 0.f), 99.f);
      int p2 = (int)fminf(fmaxf(floorf(100.f*bb), 0.f), 99.f);
      part = (int)parts3d[(p0*100 + p1)*100 + p2];
    }
    out[b*PIX + p] = part;
  }
}

extern "C" void kernel_launch(void* const* d_in, const int* in_sizes, int n_in,
                              void* d_out, int out_size, void* d_ws, size_t ws_size,
                              hipStream_t stream) {
  (void)in_sizes; (void)n_in; (void)out_size; (void)ws_size;
  const float* vertices = (const float*)d_in[0];
  const float* cam_ts   = (const float*)d_in[1];
  const int*   faces    = (const int*)  d_in[2];
  const float* textures = (const float*)d_in[3];
  const float* cube     = (const float*)d_in[4];
  const float* camK     = (const float*)d_in[5];
  const float* camR     = (const float*)d_in[6];
  int*   out  = (int*)d_out;
  float* coef = (float*)d_ws;   // NB*NFACE*REC floats ≈ 1.7 MB

  nmr_setup<<<(NB*NFACE + 255)/256, 256, 0, stream>>>(vertices, cam_ts, faces, camK, camR, coef);
  nmr_raster<<<dim3(PIX/128, NB), 256, 0, stream>>>(coef, textures, cube, out);
}